// CTCLTH_log_16896401342528
// MI455X (gfx1250) — compile-verified
//
#include <hip/hip_runtime.h>

#ifndef __has_builtin
#define __has_builtin(x) 0
#endif

// Problem constants (match reference exactly)
#define CTC_T 2048
#define CTC_N 256
#define CTC_C 128
#define CTC_L 200
#define CTC_S (2 * CTC_L + 1)   // 401 padded-label states
#define BLOCK 416               // 13 wave32s >= S

#define LOG2E 1.4426950408889634f
#define LN2   0.6931471805599453f
// NEG surrogate, pre-scaled into the log2 domain (reference NEG = -1e30 nats)
#define NEG2F (-1.0e30f * LOG2E)

#if defined(__gfx1250__) && __has_builtin(__builtin_amdgcn_global_load_async_to_lds_b128)
#define HAVE_ASYNC_LDS 1
#else
#define HAVE_ASYNC_LDS 0
#endif

// Builtin signature (from hipcc diagnostic): (v4i AS1*, v4i AS3*, imm int, imm int)
typedef int v4i __attribute__((vector_size(16)));
typedef __attribute__((address_space(1))) v4i* gas1_v4i;
typedef __attribute__((address_space(3))) v4i* gas3_v4i;

__device__ __forceinline__ void wait_asynccnt0() {
#if __has_builtin(__builtin_amdgcn_s_wait_asynccnt)
  __builtin_amdgcn_s_wait_asynccnt(0);
#else
  asm volatile("s_wait_asynccnt 0x0" ::: "memory");
#endif
}

// Raw hardware base-2 transcendentals (v_exp_f32 / v_log_f32)
__device__ __forceinline__ float exp2_hw(float x) { return __builtin_amdgcn_exp2f(x); }
__device__ __forceinline__ float log2_hw(float x) { return __builtin_amdgcn_logf(x); }

// One workgroup per sample n. Thread s owns CTC state s.
// The whole alpha recursion runs scaled by log2(e): ln-domain logsumexp
// commutes with uniform scaling, so it becomes an exact log2-domain
// logsumexp using the native base-2 exp/log units (no ln<->log2 muls on
// the 2048-step serial critical path).
__global__ void __launch_bounds__(BLOCK)
ctc_alpha_kernel(const float* __restrict__ pred,          // (T, N, C) log-probs (nats)
                 const int*   __restrict__ target,        // (N*L,)
                 const int*   __restrict__ target_length, // (N,)
                 float*       __restrict__ loss_ws)       // (N,) per-sample loss out
{
  const int n   = blockIdx.x;
  const int tid = threadIdx.x;

  __shared__ float predbuf[2][CTC_C];      // double-buffered pred rows (512 B each, raw nats)
  __shared__ float alpha[2][CTC_S + 2];    // double-buffered alpha (log2 domain), 2-entry NEG pad

  // ---- Loop-invariant per-thread state (registers): ext[s] label + skip flag ----
  // Reference: start_n = (n==0) ? 0 : target_length[n-1]
  const int start = (n == 0) ? 0 : target_length[n - 1];
  int  lab  = 0;       // ext[s]: 0 (blank) at even s, target label at odd s
  bool skip = false;   // s-2 -> s transition allowed
  if (tid < CTC_S) {
    if (tid & 1) {
      lab = target[start + ((tid - 1) >> 1)];
      if (tid >= 2)
        skip = (lab != target[start + ((tid - 3) >> 1)]);
    }
  }

  // ---- alpha[t=0] into buffer 0 (scaled into log2 domain) ----
  const float* pred_row0 = pred + (size_t)n * CTC_C;   // pred[0, n, :]
  if (tid < CTC_S) {
    float a = NEG2F;
    if (tid == 0)      a = pred_row0[0]   * LOG2E;   // blank
    else if (tid == 1) a = pred_row0[lab] * LOG2E;   // first label
    alpha[0][2 + tid] = a;
  }
  if (tid < 2) { alpha[0][tid] = NEG2F; alpha[1][tid] = NEG2F; }
  __syncthreads();

  const size_t rowstride = (size_t)CTC_N * CTC_C;

  // ---- Prefetch pred row t=1 into predbuf[1] via the async data mover ----
  {
    const float* row1 = pred + rowstride + (size_t)n * CTC_C;
#if HAVE_ASYNC_LDS
    if (tid < 32) {
      __builtin_amdgcn_global_load_async_to_lds_b128(
          (gas1_v4i)(row1 + tid * 4),
          (gas3_v4i)(&predbuf[1][tid * 4]), 0, 0);
    }
#else
    if (tid < CTC_C) predbuf[1][tid] = row1[tid];
#endif
  }

  // ---- Sequential scan over T: one barrier per step ----
  for (int t = 1; t < CTC_T; ++t) {
    const int cur = t & 1;       // pred row t lives in predbuf[cur]; write alpha[cur]
    const int prv = cur ^ 1;     // read alpha[prv]

#if HAVE_ASYNC_LDS
    if (tid < 32) wait_asynccnt0();    // row t has landed in LDS (wave 0 issued it)
#endif
    __syncthreads();                    // row t visible + alpha[prv] complete everywhere

    // Prefetch row t+1 into the buffer the compute below does NOT touch.
    if (t + 1 < CTC_T) {
      const float* rowN = pred + (size_t)(t + 1) * rowstride + (size_t)n * CTC_C;
#if HAVE_ASYNC_LDS
      if (tid < 32) {
        __builtin_amdgcn_global_load_async_to_lds_b128(
            (gas1_v4i)(rowN + tid * 4),
            (gas3_v4i)(&predbuf[prv][tid * 4]), 0, 0);
      }
#else
      if (tid < CTC_C) predbuf[prv][tid] = rowN[tid];
#endif
    }

    if (tid < CTC_S) {
      const float a0 = alpha[prv][2 + tid];
      const float a1 = alpha[prv][1 + tid];
      const float a2 = skip ? alpha[prv][tid] : NEG2F;
      const float m  = fmaxf(a0, fmaxf(a1, a2));               // v_max3
      const float s  = exp2_hw(a0 - m) + exp2_hw(a1 - m) + exp2_hw(a2 - m);
      const float sc = predbuf[cur][lab] * LOG2E;              // emission gather, nats -> log2
      alpha[cur][2 + tid] = m + log2_hw(s) + sc;
    }
    // next iteration's top barrier separates this write from its readers
  }

  __syncthreads();
  if (tid == 0) {
    const int fin = (CTC_T - 1) & 1;
    const float aS2 = alpha[fin][2 + CTC_S - 2];
    const float aS1 = alpha[fin][2 + CTC_S - 1];
    const float m   = fmaxf(aS2, aS1);
    // unscale log2 domain -> nats once, at the very end
    loss_ws[n] = -LN2 * (m + log2_hw(exp2_hw(aS2 - m) + exp2_hw(aS1 - m)));
  }
}

// Deterministic fixed-order reduction of the N per-sample losses.
__global__ void ctc_reduce_kernel(const float* __restrict__ loss_ws,
                                  float* __restrict__ out)
{
  if (blockIdx.x == 0 && threadIdx.x == 0) {
    float s = 0.0f;
    for (int i = 0; i < CTC_N; ++i) s += loss_ws[i];
    out[0] = s;
  }
}

extern "C" void kernel_launch(void* const* d_in, const int* in_sizes, int n_in,
                              void* d_out, int out_size, void* d_ws, size_t ws_size,
                              hipStream_t stream) {
  const float* pred          = (const float*)d_in[0];  // (T,N,C) f32 log-probs
  const int*   target        = (const int*)d_in[1];    // (N*L,) i32
  // d_in[2] = input_length (unused by reference math; all == T)
  const int*   target_length = (const int*)d_in[3];    // (N,) i32
  float*       out           = (float*)d_out;          // scalar f32
  float*       loss_ws       = (float*)d_ws;           // N floats scratch

  ctc_alpha_kernel<<<dim3(CTC_N), dim3(BLOCK), 0, stream>>>(
      pred, target, target_length, loss_ws);
  ctc_reduce_kernel<<<dim3(1), dim3(32), 0, stream>>>(loss_ws, out);
}